// GraphConv_6648609374671
// MI455X (gfx1250) — compile-verified
//
#include <hip/hip_runtime.h>

typedef __attribute__((ext_vector_type(2))) float v2f;
typedef __attribute__((ext_vector_type(4))) float v4f;
typedef __attribute__((ext_vector_type(8))) float v8f;

#define D_IN  256
#define D_OUT 64

// ---------------------------------------------------------------------------
// Kernel 0: zero the accumulation target (d_out), float4-vectorized.
// ---------------------------------------------------------------------------
__global__ __launch_bounds__(256) void k_zero(float* __restrict__ out, int n4) {
    int i = blockIdx.x * 256 + threadIdx.x;
    if (i < n4) {
        v4f z = {0.f, 0.f, 0.f, 0.f};
        ((v4f*)out)[i] = z;
    }
}

// ---------------------------------------------------------------------------
// Kernel 1: hidden = x @ W using V_WMMA_F32_16X16X4_F32.
//   - W (256x64 = 64 KB) staged into LDS, pre-swizzled so that for a K-pair
//     (k, k+1) and column n the two values sit adjacent:
//        wl[(k>>1)*128 + n*2 + (k&1)]
//     -> each lane fetches its B fragment with a single ds_load_b64.
//   - One wave computes a 16x64 output tile: 4 accumulators (v8f each),
//     K loop of 64 steps x 4 WMMAs.
//   A-fragment layout (32-bit A 16x4): lanes 0-15 hold K={k,k+1} for row M=lane,
//   lanes 16-31 hold K={k+2,k+3} for row M=lane-16  -> one v2f global load.
//   B-fragment (4x16): VGPR0/1 = K rows striped across lanes, same half split.
//   D layout (32-bit C/D 16x16): VGPR r -> M = r + 8*(lane>=16), N = lane&15.
// ---------------------------------------------------------------------------
__global__ __launch_bounds__(256) void k_gemm(const float* __restrict__ x,
                                              const float* __restrict__ w,
                                              float* __restrict__ hidden,
                                              int n_rows) {
    __shared__ float wl[D_IN * D_OUT];   // 64 KB (WGP has 320 KB LDS)

    const int tid = threadIdx.x;
    // Cooperative swizzled stage of W: element (k, n) -> wl[(k/2)*128 + n*2 + k%2]
    for (int i = tid; i < D_IN * D_OUT; i += 256) {
        int k = i / D_OUT;
        int n = i - k * D_OUT;
        wl[(k >> 1) * 128 + n * 2 + (k & 1)] = w[i];
    }
    __syncthreads();

    const int wave = tid >> 5;
    const int lane = tid & 31;
    const int l15  = lane & 15;
    const int hi   = lane >> 4;          // 0: K{0,1}, 1: K{2,3} half

    const long row0 = ((long)blockIdx.x * 8 + wave) * 16;
    if (row0 >= n_rows) return;          // wave-uniform: EXEC stays all-ones

    const float* ap = x + (row0 + l15) * D_IN + 2 * hi;

    v8f c0 = {}, c1 = {}, c2 = {}, c3 = {};

    for (int k = 0; k < D_IN; k += 4) {
        v2f a = *(const v2f*)(ap + k);
        const float* bbase = &wl[((k >> 1) + hi) * 128 + l15 * 2];
        v2f b0 = *(const v2f*)(bbase +  0);   // cols  0..15
        v2f b1 = *(const v2f*)(bbase + 32);   // cols 16..31
        v2f b2 = *(const v2f*)(bbase + 64);   // cols 32..47
        v2f b3 = *(const v2f*)(bbase + 96);   // cols 48..63
        c0 = __builtin_amdgcn_wmma_f32_16x16x4_f32(false, a, false, b0, (short)0, c0, false, false);
        c1 = __builtin_amdgcn_wmma_f32_16x16x4_f32(false, a, false, b1, (short)0, c1, false, false);
        c2 = __builtin_amdgcn_wmma_f32_16x16x4_f32(false, a, false, b2, (short)0, c2, false, false);
        c3 = __builtin_amdgcn_wmma_f32_16x16x4_f32(false, a, false, b3, (short)0, c3, false, false);
    }

    // Write back: VGPR r holds row (row0 + r + 8*hi), column (tile*16 + l15).
    float* hp = hidden + (row0 + 8 * hi) * D_OUT + l15;
#pragma unroll
    for (int r = 0; r < 8; ++r) {
        hp[r * D_OUT +  0] = c0[r];
        hp[r * D_OUT + 16] = c1[r];
        hp[r * D_OUT + 32] = c2[r];
        hp[r * D_OUT + 48] = c3[r];
    }
}

// ---------------------------------------------------------------------------
// Kernel 2: edge scatter. One wave per edge:
//   out[row[e]][:] += adj[e] * hidden[col[e]][:]
// 64 contiguous f32 loads + 64 contiguous global_atomic_add_f32 per edge.
// hidden (25.6 MB) and out (25.6 MB) both fit in the 192 MB L2, so the
// random-row gathers and the RMW atomics are L2-resident.
// ---------------------------------------------------------------------------
__global__ __launch_bounds__(256) void k_scatter(const float* __restrict__ hidden,
                                                 const float* __restrict__ adj,
                                                 const int* __restrict__ row,
                                                 const int* __restrict__ col,
                                                 float* __restrict__ out,
                                                 int n_edges) {
    const int wave = threadIdx.x >> 5;
    const int lane = threadIdx.x & 31;
    const int e    = blockIdx.x * 8 + wave;
    if (e >= n_edges) return;

    const int   r = row[e];
    const int   c = col[e];
    const float v = adj[e];

    const float* h = hidden + (long)c * D_OUT;
    float*       o = out    + (long)r * D_OUT;

    atomicAdd(&o[lane],      v * h[lane]);
    atomicAdd(&o[lane + 32], v * h[lane + 32]);
}

// ---------------------------------------------------------------------------
// Kernel 3: PReLU in place on d_out (single shared slope).
// ---------------------------------------------------------------------------
__global__ __launch_bounds__(256) void k_prelu(float* __restrict__ out,
                                               const float* __restrict__ slope,
                                               int n4) {
    int i = blockIdx.x * 256 + threadIdx.x;
    if (i >= n4) return;
    const float s = slope[0];
    v4f v = ((v4f*)out)[i];
#pragma unroll
    for (int j = 0; j < 4; ++j) {
        float t = v[j];
        v[j] = t > 0.f ? t : s * t;
    }
    ((v4f*)out)[i] = v;
}

// ---------------------------------------------------------------------------
// Launcher
// inputs: x[100000*256] f32, weight[256*64] f32, adj_vals[1e6] f32,
//         prelu_a[1] f32, row[1e6] i32, col[1e6] i32
// output: [100000*64] f32. Workspace: hidden (25.6 MB) lives in d_ws.
// ---------------------------------------------------------------------------
extern "C" void kernel_launch(void* const* d_in, const int* in_sizes, int n_in,
                              void* d_out, int out_size, void* d_ws, size_t ws_size,
                              hipStream_t stream) {
    const float* x    = (const float*)d_in[0];
    const float* w    = (const float*)d_in[1];
    const float* adj  = (const float*)d_in[2];
    const float* pa   = (const float*)d_in[3];
    const int*   row  = (const int*)d_in[4];
    const int*   col  = (const int*)d_in[5];
    float*       out  = (float*)d_out;
    float*       hid  = (float*)d_ws;          // n_nodes * 64 floats

    const int n_nodes = in_sizes[0] / D_IN;    // 100000
    const int n_edges = in_sizes[2];           // 1000000
    const int out4    = out_size / 4;

    // 1) zero accumulation target
    k_zero<<<(out4 + 255) / 256, 256, 0, stream>>>(out, out4);

    // 2) hidden = x @ W   (one 16-row tile per wave, 8 waves per block)
    const int row_tiles = (n_nodes + 15) / 16;
    k_gemm<<<(row_tiles + 7) / 8, 256, 0, stream>>>(x, w, hid, n_nodes);

    // 3) edge scatter with f32 atomics (one wave per edge)
    k_scatter<<<(n_edges + 7) / 8, 256, 0, stream>>>(hid, adj, row, col, out, n_edges);

    // 4) PReLU in place
    k_prelu<<<(out4 + 255) / 256, 256, 0, stream>>>(out, pa, out4);
}